// firing_model_74801150427210
// MI455X (gfx1250) — compile-verified
//
#include <hip/hip_runtime.h>
#include <math.h>

// CDNA5 / gfx1250: wave32, WMMA 16x16x4 f32.
typedef __attribute__((ext_vector_type(2))) float v2f;
typedef __attribute__((ext_vector_type(8))) float v8f;

#define SIZE 69
#define NT   5     // N padded to 80 = 5 tiles of 16
#define KC   18    // K padded to 72 = 18 chunks of 4

// Single-wave persistent kernel: runs the whole 499,799-step recurrence.
//   next = tanh(prev @ x) - prev
// prev lives in LDS (80 floats, zero-padded); x is pre-swizzled once into
// per-tile WMMA B-layout in LDS. Each step does 90 v_wmma_f32_16x16x4_f32
// (18 K-chunks x 5 N-tiles) with A = prev chunk broadcast to all 16 rows,
// so every row of D equals the GEMV result; lane L reads y[16n + (L&15)]
// from acc[0] (duplicated in both half-waves per the C/D layout).
__global__ __launch_bounds__(32)
void firing_model_74801150427210_kernel(const float* __restrict__ x,
                                        float* __restrict__ out,
                                        int rows) {
    // B tiles: tile (k,n) occupies 32 lanes x 8B. Lane L holds
    // { W[4k + 2*(L>=16)][16n + (L&15)], W[4k + 2*(L>=16) + 1][16n + (L&15)] }
    __shared__ v2f Wt[KC * NT * 32];
    __shared__ v2f prevV[40];          // 80 floats, zero-padded state

    const int l     = threadIdx.x;     // 0..31 (wave32)
    const int hi    = (l >= 16) ? 1 : 0;
    const int col15 = l & 15;

    // ---- one-time setup: swizzle x into WMMA B-tile layout (zero padded) ----
    for (int t = 0; t < KC * NT; ++t) {
        int k  = t / NT;
        int n  = t - k * NT;
        int r0 = 4 * k + 2 * hi;
        int c  = 16 * n + col15;
        v2f v;
        v.x = (r0     < SIZE && c < SIZE) ? x[r0 * SIZE + c]       : 0.0f;
        v.y = (r0 + 1 < SIZE && c < SIZE) ? x[(r0 + 1) * SIZE + c] : 0.0f;
        Wt[t * 32 + l] = v;
    }
    for (int j = l; j < 40; j += 32) {
        v2f z; z.x = 0.0f; z.y = 0.0f;
        prevV[j] = z;
    }
    __syncthreads();  // single-wave workgroup: lowers to s_nop; LDS is in-order per wave

    // per-lane copy of prev[16n + (L&15)] (consistent across both half-waves)
    float pv[NT];
#pragma unroll
    for (int n = 0; n < NT; ++n) pv[n] = 0.0f;

    // row 0 = init = zeros
    if (l < 16) {
#pragma unroll
        for (int n = 0; n < NT; ++n) {
            int j = 16 * n + l;
            if (j < SIZE) out[j] = 0.0f;
        }
    }

    // ---- serial time loop ----
    for (int i = 1; i < rows; ++i) {
        // A chunks: broadcast prev[4k..4k+3] to all 16 rows.
        // ISA A-layout (16x4 f32): lanes 0-15 -> {K=0,K=1}, lanes 16-31 -> {K=2,K=3}.
        v2f a[KC];
#pragma unroll
        for (int k = 0; k < KC; ++k) a[k] = prevV[2 * k + hi];

        v8f acc[NT];
#pragma unroll
        for (int n = 0; n < NT; ++n) {
            v8f z = {0.f, 0.f, 0.f, 0.f, 0.f, 0.f, 0.f, 0.f};
            acc[n] = z;
        }

#pragma unroll
        for (int n = 0; n < NT; ++n) {
#pragma unroll
            for (int k = 0; k < KC; ++k) {
                acc[n] = __builtin_amdgcn_wmma_f32_16x16x4_f32(
                    /*neg_a=*/false, a[k],
                    /*neg_b=*/false, Wt[(k * NT + n) * 32 + l],
                    /*c_mod=*/(short)0, acc[n],
                    /*reuse_a=*/false, /*reuse_b=*/false);
            }
        }

        // next = tanh(y) - prev ; padded lanes stay exactly 0 (tanh(0)-0).
#pragma unroll
        for (int n = 0; n < NT; ++n) {
            float y  = acc[n][0];          // D row M=0 (lanes 0-15) == row M=8 (lanes 16-31)
            float nx = tanhf(y) - pv[n];
            pv[n] = nx;
            if (l < 16) {
                ((__shared__ float*)prevV)[16 * n + l] = nx;
                int j = 16 * n + l;
                if (j < SIZE) out[(size_t)i * SIZE + j] = nx;
            }
        }
    }
}

extern "C" void kernel_launch(void* const* d_in, const int* in_sizes, int n_in,
                              void* d_out, int out_size, void* d_ws, size_t ws_size,
                              hipStream_t stream) {
    (void)in_sizes; (void)n_in; (void)d_ws; (void)ws_size;
    const float* x = (const float*)d_in[0];   // 69x69 weight matrix
    // d_in[1] (time_fomer) and d_in[2] (time) do not affect the returned rows.
    float* out = (float*)d_out;
    int rows = out_size / SIZE;               // = TIME - START = 499800
    firing_model_74801150427210_kernel<<<1, 32, 0, stream>>>(x, out, rows);
}